// GCN_Classic_21449066676828
// MI455X (gfx1250) — compile-verified
//
#include <hip/hip_runtime.h>

typedef float v2f __attribute__((ext_vector_type(2)));
typedef float v8f __attribute__((ext_vector_type(8)));

#define F_CH 128
#define EPS_BN 1e-5f

// ---------------- utility kernels ----------------

__global__ void k_fill(float* p, float v, int n) {
    int i = blockIdx.x * blockDim.x + threadIdx.x;
    if (i < n) p[i] = v;
}

__global__ void k_deg(const long long* __restrict__ col, float* __restrict__ deg, int E) {
    int e = blockIdx.x * blockDim.x + threadIdx.x;
    if (e < E) atomicAdd(&deg[col[e]], 1.0f);
}

__global__ void k_dinv(float* deg, int n) {
    int i = blockIdx.x * blockDim.x + threadIdx.x;
    if (i < n) {
        float d = deg[i];
        deg[i] = (d > 0.0f) ? rsqrtf(d) : 0.0f;
    }
}

__global__ void k_norm(const long long* __restrict__ row, const long long* __restrict__ col,
                       const float* __restrict__ dinv, float* __restrict__ norm, int E) {
    int e = blockIdx.x * blockDim.x + threadIdx.x;
    if (e < E) norm[e] = dinv[row[e]] * dinv[col[e]];
}

__global__ void k_init_bias(float* __restrict__ h, const float* __restrict__ bias, int total) {
    int i = blockIdx.x * blockDim.x + threadIdx.x;
    if (i < total) h[i] = bias[i & (F_CH - 1)];
}

// ---------------- WMMA GEMM: [M,128] x [128,128], W staged in LDS ----------------
// LDS holds W k-pair interleaved: w2[p*128 + n] = {W[2p][n], W[2p+1][n]}
// -> each B fragment is one aligned 8-byte LDS load (no repacking movs).
// Each wave computes TWO 16-row M tiles (32 rows) so every B fragment feeds 2 WMMAs.
__global__ __launch_bounds__(256) void k_gemm128_lds(const float* __restrict__ A,
                                                     const float* __restrict__ W,
                                                     float* __restrict__ C, int M) {
    __shared__ v2f w2[64 * F_CH];   // 64 KB
    int tid = threadIdx.x;
    #pragma unroll
    for (int i = 0; i < 32; ++i) {
        int idx = tid + i * 256;            // 0..8191 : p = idx>>7, n = idx&127
        int p = idx >> 7, n = idx & 127;
        v2f t;
        t[0] = W[(size_t)(2 * p) * F_CH + n];
        t[1] = W[(size_t)(2 * p + 1) * F_CH + n];
        w2[idx] = t;
    }
    __syncthreads();

    int wave = tid >> 5, lane = tid & 31;
    int tile0 = blockIdx.x * 256 + wave * 32;
    if (tile0 >= M) return;                  // wave-uniform: EXEC all-1s for WMMA
    int two = (tile0 + 16) < M;              // wave-uniform

    int m0 = tile0 + (lane & 15);
    int m1 = two ? (m0 + 16) : m0;           // duplicate work if single tile (store guarded)
    int khalf = (lane >> 4) << 1;            // 0 | 2
    int ncol  = lane & 15;

    v8f zero = {0.f, 0.f, 0.f, 0.f, 0.f, 0.f, 0.f, 0.f};
    v8f acc0[8], acc1[8];
    #pragma unroll
    for (int nt = 0; nt < 8; ++nt) { acc0[nt] = zero; acc1[nt] = zero; }

    for (int k0 = 0; k0 < F_CH; k0 += 4) {
        int p = (k0 + khalf) >> 1;           // even kb -> pair index
        v2f a0 = *(const v2f*)(A + (size_t)m0 * F_CH + k0 + khalf);
        v2f a1 = *(const v2f*)(A + (size_t)m1 * F_CH + k0 + khalf);
        #pragma unroll
        for (int nt = 0; nt < 8; ++nt) {
            v2f b = w2[p * F_CH + nt * 16 + ncol];
            acc0[nt] = __builtin_amdgcn_wmma_f32_16x16x4_f32(
                false, a0, false, b, (short)0, acc0[nt], false, false);
            acc1[nt] = __builtin_amdgcn_wmma_f32_16x16x4_f32(
                false, a1, false, b, (short)0, acc1[nt], false, false);
        }
    }

    int rb0 = tile0 + ((lane >> 4) << 3);
    #pragma unroll
    for (int nt = 0; nt < 8; ++nt) {
        #pragma unroll
        for (int r = 0; r < 8; ++r) {
            C[(size_t)(rb0 + r) * F_CH + nt * 16 + ncol] = acc0[nt][r];
        }
    }
    if (two) {
        int rb1 = rb0 + 16;
        #pragma unroll
        for (int nt = 0; nt < 8; ++nt) {
            #pragma unroll
            for (int r = 0; r < 8; ++r) {
                C[(size_t)(rb1 + r) * F_CH + nt * 16 + ncol] = acc1[nt][r];
            }
        }
    }
}

// ---------------- WMMA GEMM: [M,K] x [K,128] + bias, B from global (head) ----------------
__global__ __launch_bounds__(256) void k_gemm_global(const float* __restrict__ A,
                                                     const float* __restrict__ B,
                                                     const float* __restrict__ bias,
                                                     float* __restrict__ C, int M, int K) {
    int tid = threadIdx.x;
    int wave = tid >> 5, lane = tid & 31;
    int tileM = blockIdx.x * 128 + wave * 16;
    if (tileM >= M) return;

    int mrow  = tileM + (lane & 15);
    int khalf = (lane >> 4) << 1;
    int ncol  = lane & 15;

    v8f zero = {0.f, 0.f, 0.f, 0.f, 0.f, 0.f, 0.f, 0.f};
    v8f acc[8];
    #pragma unroll
    for (int nt = 0; nt < 8; ++nt) acc[nt] = zero;

    for (int k0 = 0; k0 < K; k0 += 4) {
        v2f a = *(const v2f*)(A + (size_t)mrow * K + k0 + khalf);
        const float* brow = B + (size_t)(k0 + khalf) * F_CH;
        #pragma unroll
        for (int nt = 0; nt < 8; ++nt) {
            v2f b;
            b[0] = brow[nt * 16 + ncol];
            b[1] = brow[F_CH + nt * 16 + ncol];
            acc[nt] = __builtin_amdgcn_wmma_f32_16x16x4_f32(
                false, a, false, b, (short)0, acc[nt], false, false);
        }
    }

    int rbase = tileM + ((lane >> 4) << 3);
    #pragma unroll
    for (int nt = 0; nt < 8; ++nt) {
        float bv = bias[nt * 16 + ncol];
        #pragma unroll
        for (int r = 0; r < 8; ++r) {
            C[(size_t)(rbase + r) * F_CH + nt * 16 + ncol] = acc[nt][r] + bv;
        }
    }
}

// ---------------- edge gather/scatter: one wave32 per edge pair, prefetch next ----------------
__global__ __launch_bounds__(256) void k_scatter(const float* __restrict__ hw,
                                                 const long long* __restrict__ row,
                                                 const long long* __restrict__ col,
                                                 const float* __restrict__ norm,
                                                 float* __restrict__ out, int E) {
    int wid  = blockIdx.x * 8 + (threadIdx.x >> 5);
    int lane = threadIdx.x & 31;
    int e0 = wid * 2;
    if (e0 >= E) return;
    int e1 = e0 + 1;
    int has1 = e1 < E;                        // wave-uniform

    long long r0 = row[e0];
    long long c0 = col[e0];
    float n0 = norm[e0];
    long long r1 = 0, c1 = 0;
    float n1 = 0.f;
    if (has1) {
        r1 = row[e1]; c1 = col[e1]; n1 = norm[e1];
        __builtin_prefetch(hw + (size_t)r1 * F_CH + lane * 4, 0, 3);  // global_prefetch_b8
    }

    float4 v0 = ((const float4*)(hw + (size_t)r0 * F_CH))[lane];
    float* d0 = out + (size_t)c0 * F_CH + lane * 4;
    atomicAdd(d0 + 0, v0.x * n0);
    atomicAdd(d0 + 1, v0.y * n0);
    atomicAdd(d0 + 2, v0.z * n0);
    atomicAdd(d0 + 3, v0.w * n0);

    if (has1) {
        float4 v1 = ((const float4*)(hw + (size_t)r1 * F_CH))[lane];
        float* d1 = out + (size_t)c1 * F_CH + lane * 4;
        atomicAdd(d1 + 0, v1.x * n1);
        atomicAdd(d1 + 1, v1.y * n1);
        atomicAdd(d1 + 2, v1.z * n1);
        atomicAdd(d1 + 3, v1.w * n1);
    }
}

// ---------------- relu (optional) + per-channel sum/sumsq ----------------
__global__ __launch_bounds__(128) void k_stats(float* __restrict__ h, float* __restrict__ sum,
                                               float* __restrict__ sumsq, int rows, int do_relu) {
    int f = threadIdx.x;
    int r0 = blockIdx.x * 64;
    float s = 0.f, s2 = 0.f;
    for (int i = 0; i < 64; ++i) {
        int r = r0 + i;
        if (r >= rows) break;
        float v = h[(size_t)r * F_CH + f];
        if (do_relu) { v = fmaxf(v, 0.f); h[(size_t)r * F_CH + f] = v; }
        s += v;
        s2 += v * v;
    }
    atomicAdd(&sum[f], s);
    atomicAdd(&sumsq[f], s2);
}

__global__ void k_bn_final(const float* __restrict__ sum, const float* __restrict__ sumsq,
                           float* __restrict__ murs, float rows) {
    int f = threadIdx.x;
    float mu = sum[f] / rows;
    float var = sumsq[f] / rows - mu * mu;
    murs[f] = mu;
    murs[F_CH + f] = rsqrtf(var + EPS_BN);
}

// BN apply (in-place) + graph pooling scatter into z[G, L*F]
__global__ void k_bn_pool(float* __restrict__ h, const float* __restrict__ murs,
                          const float* __restrict__ gamma, const float* __restrict__ beta,
                          const long long* __restrict__ batch, float* __restrict__ z,
                          int rows, int layer) {
    int i = blockIdx.x * blockDim.x + threadIdx.x;
    if (i >= rows * F_CH) return;
    int r = i >> 7, f = i & (F_CH - 1);
    float v = (h[i] - murs[f]) * murs[F_CH + f] * gamma[f] + beta[f];
    h[i] = v;
    atomicAdd(&z[(size_t)batch[r] * (4 * F_CH) + layer * F_CH + f], v);
}

// head: BN then relu
__global__ void k_bn_relu(const float* __restrict__ z1, const float* __restrict__ murs,
                          const float* __restrict__ g, const float* __restrict__ b,
                          float* __restrict__ z2, int n) {
    int i = blockIdx.x * blockDim.x + threadIdx.x;
    if (i >= n) return;
    int f = i & (F_CH - 1);
    float v = (z1[i] - murs[f]) * murs[F_CH + f] * g[f] + b[f];
    z2[i] = fmaxf(v, 0.f);
}

// final tiny projection [G,128] x [128,10] + b2
__global__ void k_head(const float* __restrict__ z2, const float* __restrict__ W2,
                       const float* __restrict__ b2, float* __restrict__ out, int G, int OUT) {
    int i = blockIdx.x * blockDim.x + threadIdx.x;
    if (i >= G * OUT) return;
    int g = i / OUT, o = i % OUT;
    float acc = b2[o];
    #pragma unroll 8
    for (int k = 0; k < F_CH; ++k) acc += z2[(size_t)g * F_CH + k] * W2[(size_t)k * OUT + o];
    out[i] = acc;
}

// ---------------- host launch ----------------

extern "C" void kernel_launch(void* const* d_in, const int* in_sizes, int n_in,
                              void* d_out, int out_size, void* d_ws, size_t ws_size,
                              hipStream_t stream) {
    const float*     x      = (const float*)d_in[0];
    const long long* edge   = (const long long*)d_in[1];
    const long long* batch  = (const long long*)d_in[2];
    const float*     Ws     = (const float*)d_in[3];
    const float*     bs     = (const float*)d_in[4];
    const float*     gammas = (const float*)d_in[5];
    const float*     betas  = (const float*)d_in[6];
    const float*     W1     = (const float*)d_in[7];
    const float*     b1     = (const float*)d_in[8];
    const float*     g1     = (const float*)d_in[9];
    const float*     bt1    = (const float*)d_in[10];
    const float*     W2     = (const float*)d_in[11];
    const float*     b2     = (const float*)d_in[12];
    float* out = (float*)d_out;

    const int F = F_CH, L = 4, OUT = 10;
    const int N = in_sizes[0] / F;
    const int E = in_sizes[1] / 2;
    const int G = out_size / OUT;
    const long long* row = edge;
    const long long* col = edge + E;

    float* ws = (float*)d_ws;
    size_t off = 0;
    float* hA   = ws + off; off += (size_t)N * F;
    float* hB   = ws + off; off += (size_t)N * F;
    float* dinv = ws + off; off += (size_t)N;  off = (off + 15) & ~(size_t)15;
    float* nrm  = ws + off; off += (size_t)E;
    float* z    = ws + off; off += (size_t)G * F * L;
    float* z1   = ws + off; off += (size_t)G * F;
    float* z2   = ws + off; off += (size_t)G * F;
    float* sum  = ws + off; off += F;          // sum and sumsq contiguous (zeroed together)
    float* sumsq= ws + off; off += F;
    float* murs = ws + off; off += 2 * F;

    // GCN normalization
    k_fill<<<(N + 255) / 256, 256, 0, stream>>>(dinv, 0.f, N);
    k_deg<<<(E + 255) / 256, 256, 0, stream>>>(col, dinv, E);
    k_dinv<<<(N + 255) / 256, 256, 0, stream>>>(dinv, N);
    k_norm<<<(E + 255) / 256, 256, 0, stream>>>(row, col, dinv, nrm, E);

    // pooled output accumulator
    k_fill<<<(G * F * L + 255) / 256, 256, 0, stream>>>(z, 0.f, G * F * L);

    // h0 = x
    hipMemcpyAsync(hA, x, (size_t)N * F * sizeof(float), hipMemcpyDeviceToDevice, stream);

    for (int t = 0; t < L; ++t) {
        // hw = h @ W[t]   (fp32 WMMA, W in LDS, 2 M-tiles per wave)
        k_gemm128_lds<<<(N + 255) / 256, 256, 0, stream>>>(hA, Ws + (size_t)t * F * F, hB, N);
        // accumulator initialized with bias (h = segsum + b)
        k_init_bias<<<(N * F + 255) / 256, 256, 0, stream>>>(hA, bs + t * F, N * F);
        // scatter-add messages (2 edges per wave, prefetch second row)
        k_scatter<<<(E + 15) / 16, 256, 0, stream>>>(hB, row, col, nrm, hA, E);
        // relu + batchnorm stats
        k_fill<<<1, 256, 0, stream>>>(sum, 0.f, 2 * F);
        k_stats<<<(N + 63) / 64, 128, 0, stream>>>(hA, sum, sumsq, N, 1);
        k_bn_final<<<1, 128, 0, stream>>>(sum, sumsq, murs, (float)N);
        // bn apply + graph pooling
        k_bn_pool<<<(N * F + 255) / 256, 256, 0, stream>>>(hA, murs, gammas + t * F,
                                                           betas + t * F, batch, z, N, t);
    }

    // head: z[G, L*F] @ W1[L*F, F] + b1 -> z1  (fp32 WMMA)
    k_gemm_global<<<(G + 127) / 128, 256, 0, stream>>>(z, W1, b1, z1, G, L * F);
    k_fill<<<1, 256, 0, stream>>>(sum, 0.f, 2 * F);
    k_stats<<<(G + 63) / 64, 128, 0, stream>>>(z1, sum, sumsq, G, 0);
    k_bn_final<<<1, 128, 0, stream>>>(sum, sumsq, murs, (float)G);
    k_bn_relu<<<(G * F + 255) / 256, 256, 0, stream>>>(z1, murs, g1, bt1, z2, G * F);
    k_head<<<(G * OUT + 255) / 256, 256, 0, stream>>>(z2, W2, b2, out, G, OUT);
}